// SphericalConvMHF_48850958025057
// MI455X (gfx1250) — compile-verified
//
#include <hip/hip_runtime.h>
#include <math.h>

// ---------------------------------------------------------------- constants
#define NLAT 256
#define NLON 256
#define LMAX 128
#define MMAX 129
#define CIN  32
#define COUT 32
#define BATCH 8
#define NC   (2*LMAX*MMAX)        // 33024
#define BC   (BATCH*CIN)          // 256
#define ROWS_BCK (BC*NLAT)        // 65536 rows for the DFT GEMMs
#define FLD  320                  // padded DFT col count (258 -> 320, 5x64)
#define GPAD 264                  // padded mri count (258 -> 264)

// Column-major LDS panel: sB[col*SB_STRIDE + k], col in [0,64), k in [0,KCHUNK].
// SB_STRIDE % 64 == 4  =>  b64 fragment reads are perfectly bank-conflict-free:
// lanes 0-15 hit bank pairs {4L,4L+1}, lanes 16-31 (k+2) hit {4L+2,4L+3}.
#define SB_STRIDE 132
#define KCHUNK    128             // K rows staged per chunk (64*132*4 = 33 KB LDS)

typedef __attribute__((ext_vector_type(2))) float v2f;
typedef __attribute__((ext_vector_type(8))) float v8f;

__device__ __forceinline__ v8f wmma4(v2f a, v2f b, v8f c) {
  // D(16x16,f32) = A(16x4,f32) * B(4x16,f32) + C
  return __builtin_amdgcn_wmma_f32_16x16x4_f32(false, a, false, b,
                                               (short)0, c, false, false);
}

// Cooperative transposing fill: global row-major [nrow][64] (row stride ldb)
// -> LDS column-major sB[col][k]. Coalesced b128 global reads.
__device__ __forceinline__ void fill_panel(float* sB, const float* gRow0,
                                           int nrow, int ldb) {
  for (int idx = threadIdx.x; idx < nrow * 16; idx += 256) {
    int k = idx >> 4, c4 = (idx & 15) << 2;
    float4 v = *(const float4*)(gRow0 + k * ldb + c4);
    sB[(c4 + 0) * SB_STRIDE + k] = v.x;
    sB[(c4 + 1) * SB_STRIDE + k] = v.y;
    sB[(c4 + 2) * SB_STRIDE + k] = v.z;
    sB[(c4 + 3) * SB_STRIDE + k] = v.w;
  }
}

// B-fragment for col tile t at inner offset k: one ds_load_b64 straight into
// the register pair {B[k+2kh][col], B[k+2kh+1][col]} the WMMA wants.
__device__ __forceinline__ v2f frag_b(const float* sB, int t, int lrow,
                                      int k, int kh) {
  return *(const v2f*)(sB + (t * 16 + lrow) * SB_STRIDE + k + 2 * kh);
}

// ------------------------------------------------------ workspace layout (floats)
#define O_WQ    0
#define O_PCT   (O_WQ   + 256)                      // [m][l][k]   129*128*256
#define O_PCTT  (O_PCT  + 129*128*256)              // [m][k][l]
#define O_F     (O_PCTT + 129*128*256)              // [n][col]    256*320
#define O_G     (O_F    + 256*FLD)                  // [mri][n]    264*256
#define O_SPEC  (O_G    + GPAD*256)                 // [bc][x]     256*33024
#define O_OUT   (O_SPEC + BC*NC)                    // [bo][x]
#define O_XHW   (O_OUT  + BC*NC)                    // xhw [mri][bc][k] aliased w/ Xm [bok][mri]
#define WS_FLOATS (O_XHW + ROWS_BCK*GPAD)

// ================================================================ setup kernels
__global__ void k_wq(float* __restrict__ wq) {
  int j = threadIdx.x;
  double tj = M_PI * (double)j / 255.0;
  double S = 0.0;
  for (int kk = 1; kk <= 127; ++kk)
    S += cos(2.0 * tj * (double)kk) * (2.0 / (4.0 * kk * kk - 1.0));
  double c = (j == 0 || j == 255) ? 1.0 : 2.0;
  wq[j] = (float)(c / 255.0 * (1.0 - S));
}

__global__ void k_pct(float* __restrict__ pct, float* __restrict__ pctT) {
  int m = blockIdx.x;           // 0..128
  int k = threadIdx.x;          // 0..255
  double t = M_PI * (double)k / 255.0;
  double ct = cos(t), st = sin(t);
  double pmm = 0.28209479177387814;   // 1/sqrt(4*pi)
  bool valid = (m < LMAX);
  if (valid) for (int j = 1; j <= m; ++j)
    pmm *= -sqrt((2.0 * j + 1.0) / (2.0 * j)) * st;
  double p1 = 0.0, p2 = 0.0;
  for (int l = 0; l < LMAX; ++l) {
    double val = 0.0;
    if (valid && l >= m) {
      if (l == m)          val = pmm;
      else if (l == m + 1) val = sqrt(2.0 * m + 3.0) * ct * pmm;
      else {
        double dl = (double)l, dm = (double)m, dl1 = (double)(l - 1);
        double a = sqrt((4.0 * dl * dl - 1.0) / (dl * dl - dm * dm));
        double b = sqrt((dl1 * dl1 - dm * dm) / (4.0 * dl1 * dl1 - 1.0));
        val = a * (ct * p1 - b * p2);
      }
      p2 = p1; p1 = val;
    }
    pct [(m * LMAX + l) * NLAT + k] = (float)val;
    pctT[(m * NLAT + k) * LMAX + l] = (float)val;
  }
}

__global__ void k_dft(float* __restrict__ F, float* __restrict__ G) {
  int idx = blockIdx.x * 256 + threadIdx.x;
  const double W = 2.0 * M_PI / (double)NLON;
  if (idx < NLON * FLD) {                       // forward DFT table F[n][col]
    int n = idx / FLD, col = idx % FLD;
    int m = col >> 1, ri = col & 1;
    float v = 0.0f;
    if (col < 2 * MMAX) {
      double ang = W * (double)m * (double)n;
      v = (float)((ri ? -sin(ang) : cos(ang)) * W);
    }
    F[idx] = v;
  } else if (idx < NLON * FLD + GPAD * NLON) {  // inverse table G[mri][n]
    int i2 = idx - NLON * FLD;
    int mri = i2 / NLON, n = i2 % NLON;
    int m = mri >> 1, ri = mri & 1;
    float v = 0.0f;
    if (mri < 2 * MMAX) {
      double cm = (m == 0 || m == LMAX) ? 1.0 : 2.0;
      double ang = W * (double)m * (double)n;
      v = (float)(cm * (ri ? -sin(ang) : cos(ang)));
    }
    G[i2] = v;
  }
}

__global__ void k_padXm(float* __restrict__ Xm) {   // zero cols 258..263
  int idx = blockIdx.x * 256 + threadIdx.x;         // 65536*6 elements
  int r = idx / 6, c = 258 + idx % 6;
  Xm[r * GPAD + c] = 0.0f;
}

// ================================================================ Stage A: rFFT as GEMM
// rows=(bc,k) 65536, inner n=256, cols=mri (pad 320); folds quadrature wq[k].
// 8 waves/block share one 64-col F panel staged in LDS.
__global__ __launch_bounds__(256) void k_fwd_dft(
    const float* __restrict__ x, const float* __restrict__ F,
    const float* __restrict__ wq, float* __restrict__ xhw) {
  __shared__ float sB[64 * SB_STRIDE];
  int wave = threadIdx.x >> 5, lane = threadIdx.x & 31;
  int lrow = lane & 15, kh = lane >> 4;
  int cg = blockIdx.x % 5;
  int rowBlock = blockIdx.x / 5;                  // 0..511
  int rt = rowBlock * 8 + wave;                   // 0..4095
  int r0 = rt * 16, c0 = cg * 64;
  const float* Ar = x + (r0 + lrow) * NLON;
  v8f acc[4] = {v8f{}, v8f{}, v8f{}, v8f{}};
  for (int kb = 0; kb < NLON; kb += KCHUNK) {
    __syncthreads();
    fill_panel(sB, F + kb * FLD + c0, KCHUNK, FLD);
    __syncthreads();
    for (int k = 0; k < KCHUNK; k += 4) {
      v2f a = *(const v2f*)(Ar + kb + k + 2 * kh);
#pragma unroll
      for (int t = 0; t < 4; ++t)
        acc[t] = wmma4(a, frag_b(sB, t, lrow, k, kh), acc[t]);
    }
  }
#pragma unroll
  for (int j = 0; j < 8; ++j) {
    int row = r0 + j + 8 * kh;
    float w = wq[row & (NLAT - 1)];
#pragma unroll
    for (int t = 0; t < 4; ++t) {
      int col = c0 + t * 16 + lrow;
      if (col < 2 * MMAX) xhw[col * ROWS_BCK + row] = acc[t][j] * w;
    }
  }
}

// ================================================================ Stage B: forward Legendre
// per m: rows=(ri,bc) 512, inner k=256, cols l=128
__global__ __launch_bounds__(256) void k_fwd_leg(
    const float* __restrict__ xhw, const float* __restrict__ pctT,
    float* __restrict__ spec) {
  __shared__ float sB[64 * SB_STRIDE];
  int wave = threadIdx.x >> 5, lane = threadIdx.x & 31;
  int lrow = lane & 15, kh = lane >> 4;
  int m = blockIdx.x >> 3;                          // 8 blocks per m
  int b7 = blockIdx.x & 7;
  int cg = b7 & 1, rowBlock = b7 >> 1;              // waves share cg
  int rt = rowBlock * 8 + wave;                     // 0..31
  int r0 = rt * 16, c0 = cg * 64;
  int row = r0 + lrow, ri = row >> 8, bc = row & 255;
  if (c0 + 63 < m) {                                // P_l^m == 0 for l < m (uniform per block)
#pragma unroll
    for (int t = 0; t < 4; ++t)
#pragma unroll
      for (int j = 0; j < 8; ++j) {
        int ro = r0 + j + 8 * kh, ri2 = ro >> 8, bc2 = ro & 255;
        spec[bc2 * NC + (c0 + t * 16 + lrow) * (2 * MMAX) + m * 2 + ri2] = 0.0f;
      }
    return;
  }
  const float* Ar = xhw + ((2 * m + ri) * BC + bc) * NLAT;
  const float* Bb = pctT + m * NLAT * LMAX;
  v8f acc[4] = {v8f{}, v8f{}, v8f{}, v8f{}};
  for (int kb = 0; kb < NLAT; kb += KCHUNK) {
    __syncthreads();
    fill_panel(sB, Bb + kb * LMAX + c0, KCHUNK, LMAX);
    __syncthreads();
    for (int k = 0; k < KCHUNK; k += 4) {
      v2f a = *(const v2f*)(Ar + kb + k + 2 * kh);
#pragma unroll
      for (int t = 0; t < 4; ++t)
        acc[t] = wmma4(a, frag_b(sB, t, lrow, k, kh), acc[t]);
    }
  }
#pragma unroll
  for (int t = 0; t < 4; ++t) {
    int l = c0 + t * 16 + lrow;
#pragma unroll
    for (int j = 0; j < 8; ++j) {
      int ro = r0 + j + 8 * kh, ri2 = ro >> 8, bc2 = ro & 255;
      spec[bc2 * NC + l * (2 * MMAX) + m * 2 + ri2] = acc[t][j];
    }
  }
}

// ================================================================ Stage C: channel mix (VALU)
// out[b,o,x] = sum_i spec[b,i,x] * w[i,o,x]  -- weight-bandwidth bound
__global__ __launch_bounds__(256) void k_mix(
    const float* __restrict__ spec, const float* __restrict__ w,
    float* __restrict__ outw) {
  int xid = blockIdx.x * 256 + threadIdx.x;
  if (xid >= NC) return;
  for (int oc = 0; oc < 4; ++oc) {                  // 8 outputs per chunk
    float acc[BATCH][8];
#pragma unroll
    for (int b = 0; b < BATCH; ++b)
#pragma unroll
      for (int oj = 0; oj < 8; ++oj) acc[b][oj] = 0.0f;
    for (int i = 0; i < CIN; ++i) {
      if (i < CIN - 8)                              // prefetch the 135 MB weight stream
        __builtin_prefetch(&w[((i + 8) * COUT + oc * 8) * NC + xid], 0, 0);
      float s[BATCH];
#pragma unroll
      for (int b = 0; b < BATCH; ++b) s[b] = spec[(b * CIN + i) * NC + xid];
#pragma unroll
      for (int oj = 0; oj < 8; ++oj) {
        float wv = w[(i * COUT + oc * 8 + oj) * NC + xid];
#pragma unroll
        for (int b = 0; b < BATCH; ++b) acc[b][oj] += s[b] * wv;
      }
    }
#pragma unroll
    for (int b = 0; b < BATCH; ++b)
#pragma unroll
      for (int oj = 0; oj < 8; ++oj)
        outw[(b * COUT + oc * 8 + oj) * NC + xid] = acc[b][oj];
  }
}

// ================================================================ Stage D: inverse Legendre
// per m: rows=(ri,bo) 512, inner l (start at m&~3), cols k=256
__global__ __launch_bounds__(256) void k_inv_leg(
    const float* __restrict__ outw, const float* __restrict__ pct,
    float* __restrict__ Xm) {
  __shared__ float sB[64 * SB_STRIDE];
  int wave = threadIdx.x >> 5, lane = threadIdx.x & 31;
  int lrow = lane & 15, kh = lane >> 4;
  int m = blockIdx.x >> 4;                          // 16 blocks per m
  int b15 = blockIdx.x & 15;
  int cg = b15 & 3, rowBlock = b15 >> 2;            // waves share cg
  int rt = rowBlock * 8 + wave;                     // 0..31
  int r0 = rt * 16, c0 = cg * 64;
  int row = r0 + lrow, ri = row >> 8, bo = row & 255;
  const float* Ar = outw + bo * NC + m * 2 + ri;    // A[row][l] = Ar[l*258]
  // stage entire 128-row P panel (includes zero rows l<m), single chunk
  __syncthreads();
  fill_panel(sB, pct + m * LMAX * NLAT + c0, LMAX, NLAT);
  __syncthreads();
  v8f acc[4] = {v8f{}, v8f{}, v8f{}, v8f{}};
  for (int l = (m & ~3); l < LMAX; l += 4) {        // skip l<m (P==0)
    int la = l + 2 * kh;
    v2f a; a.x = Ar[la * (2 * MMAX)]; a.y = Ar[(la + 1) * (2 * MMAX)];
#pragma unroll
    for (int t = 0; t < 4; ++t)
      acc[t] = wmma4(a, frag_b(sB, t, lrow, l, kh), acc[t]);
  }
#pragma unroll
  for (int t = 0; t < 4; ++t) {
    int k = c0 + t * 16 + lrow;
#pragma unroll
    for (int j = 0; j < 8; ++j) {
      int ro = r0 + j + 8 * kh, ri2 = ro >> 8, bo2 = ro & 255;
      Xm[(bo2 * NLAT + k) * GPAD + m * 2 + ri2] = acc[t][j];
    }
  }
}

// ================================================================ Stage E: irFFT as GEMM
// rows=(bo,k) 65536, inner mri (264 padded), cols n=256; writes y directly.
__global__ __launch_bounds__(256) void k_inv_dft(
    const float* __restrict__ Xm, const float* __restrict__ G,
    float* __restrict__ y) {
  __shared__ float sB[64 * SB_STRIDE];
  int wave = threadIdx.x >> 5, lane = threadIdx.x & 31;
  int lrow = lane & 15, kh = lane >> 4;
  int cg = blockIdx.x & 3;
  int rowBlock = blockIdx.x >> 2;                   // 0..511
  int rt = rowBlock * 8 + wave;                     // 0..4095
  int r0 = rt * 16, c0 = cg * 64;
  const float* Ar = Xm + (r0 + lrow) * GPAD;
  v8f acc[4] = {v8f{}, v8f{}, v8f{}, v8f{}};
  for (int kb = 0; kb < GPAD; kb += KCHUNK) {
    int nr = (GPAD - kb < KCHUNK) ? (GPAD - kb) : KCHUNK;
    __syncthreads();
    fill_panel(sB, G + kb * NLON + c0, nr, NLON);
    __syncthreads();
    for (int k = 0; k < nr; k += 4) {
      v2f a = *(const v2f*)(Ar + kb + k + 2 * kh);
#pragma unroll
      for (int t = 0; t < 4; ++t)
        acc[t] = wmma4(a, frag_b(sB, t, lrow, k, kh), acc[t]);
    }
  }
#pragma unroll
  for (int t = 0; t < 4; ++t) {
    int n = c0 + t * 16 + lrow;
#pragma unroll
    for (int j = 0; j < 8; ++j) {
      int ro = r0 + j + 8 * kh;
      y[ro * NLON + n] = acc[t][j];
    }
  }
}

// ================================================================ launcher
extern "C" void kernel_launch(void* const* d_in, const int* in_sizes, int n_in,
                              void* d_out, int out_size, void* d_ws, size_t ws_size,
                              hipStream_t stream) {
  (void)in_sizes; (void)n_in; (void)out_size;
  if (ws_size < (size_t)WS_FLOATS * sizeof(float)) return;
  const float* x = (const float*)d_in[0];
  const float* weight = (const float*)d_in[1];
  float* y  = (float*)d_out;
  float* ws = (float*)d_ws;
  float* wq   = ws + O_WQ;
  float* pct  = ws + O_PCT;
  float* pctT = ws + O_PCTT;
  float* F    = ws + O_F;
  float* G    = ws + O_G;
  float* spec = ws + O_SPEC;
  float* outw = ws + O_OUT;
  float* xhw  = ws + O_XHW;     // aliased: xhw (A,B) then Xm (D,E)
  float* Xm   = ws + O_XHW;

  k_wq     <<<1,    256, 0, stream>>>(wq);
  k_pct    <<<129,  256, 0, stream>>>(pct, pctT);
  k_dft    <<<(NLON*FLD + GPAD*NLON + 255) / 256, 256, 0, stream>>>(F, G);
  k_fwd_dft<<<2560, 256, 0, stream>>>(x, F, wq, xhw);
  k_fwd_leg<<<1032, 256, 0, stream>>>(xhw, pctT, spec);
  k_mix    <<<129,  256, 0, stream>>>(spec, weight, outw);
  k_padXm  <<<1536, 256, 0, stream>>>(Xm);
  k_inv_leg<<<2064, 256, 0, stream>>>(outw, pct, Xm);
  k_inv_dft<<<2048, 256, 0, stream>>>(Xm, G, y);
}